// HyperGraphCustomBipartiteDisenGATVAEV4_44521630990696
// MI455X (gfx1250) — compile-verified
//
#include <hip/hip_runtime.h>

// ---------------- constants (match reference) ----------------
constexpr int NU = 200000;
constexpr int NI = 100000;
constexpr int NN = NU + NI;      // 300000
constexpr int EMB = 64;
constexpr int CDIM = 32;
constexpr int NE = 2000000;
constexpr int ITERS = 3;

typedef float v2f __attribute__((ext_vector_type(2)));
typedef float v8f __attribute__((ext_vector_type(8)));

// =============================================================
// Kernel 1: u = concat([user_pref, user_emb]) @ W_merge + b
//   M = 200000 (16-row tiles, exact), N = 64, K = 128
//   One block = 128 threads = 4 waves; wave w computes N-tile w.
//   fp32 WMMA 16x16x4, A tile staged in LDS (padded rows).
// =============================================================
__global__ __launch_bounds__(128) void merge_gemm(
    const float* __restrict__ user_pref, const float* __restrict__ user_emb,
    const float* __restrict__ W, const float* __restrict__ b,
    float* __restrict__ u /* [NU][64] */)
{
    __shared__ float Xs[16 * 132];                 // 16 rows x 128, +4 pad (bank stride 4)
    const int row0 = blockIdx.x * 16;
    const int tid = threadIdx.x;

    // cooperative coalesced load: 16 rows x 32 float4 (pref first, then emb)
    for (int i = tid; i < 16 * 32; i += 128) {
        const int r = i >> 5;                      // local row
        const int q = i & 31;                      // float4 index within 128 floats
        const int row = row0 + r;
        float4 v;
        if (q < 16) v = ((const float4*)(user_pref + (size_t)row * 64))[q];
        else        v = ((const float4*)(user_emb  + (size_t)row * 64))[q - 16];
        ((float4*)(Xs + r * 132))[q] = v;
    }
    __syncthreads();

    const int wave  = tid >> 5;                    // N-tile 0..3
    const int lane  = tid & 31;
    const int mrow  = lane & 15;                   // A: M index for this lane
    const int khalf = (lane >> 4) * 2;             // A/B: K sub-offset (0 or 2)
    const int col   = wave * 16 + (lane & 15);     // B/D: N index

    v8f acc = {0.f, 0.f, 0.f, 0.f, 0.f, 0.f, 0.f, 0.f};
    #pragma unroll 4
    for (int k0 = 0; k0 < 128; k0 += 4) {
        v2f a, bb;
        a[0]  = Xs[mrow * 132 + k0 + khalf + 0];
        a[1]  = Xs[mrow * 132 + k0 + khalf + 1];
        bb[0] = W[(k0 + khalf + 0) * 64 + col];
        bb[1] = W[(k0 + khalf + 1) * 64 + col];
        acc = __builtin_amdgcn_wmma_f32_16x16x4_f32(false, a, false, bb,
                                                    (short)0, acc, false, false);
    }

    const float bias = b[col];
    const int mbase = row0 + (lane >> 4) * 8;      // D: VGPR j -> M = j (+8 upper half)
    #pragma unroll
    for (int j = 0; j < 8; ++j)
        u[(size_t)(mbase + j) * 64 + col] = acc[j] + bias;
}

// =============================================================
// Kernel 2: copy item embeddings into all_emb tail
// =============================================================
__global__ __launch_bounds__(256) void copy_items(
    const float* __restrict__ src, float* __restrict__ dst)
{
    const int t = blockIdx.x * 256 + threadIdx.x;  // float4 index
    if (t < NI * 16) ((float4*)dst)[t] = ((const float4*)src)[t];
}

// =============================================================
// Kernel 3: z[c,n,k] = all_emb[n,:] @ weights[c,:,k] + biases[c,k]
//   M = 300000 (exact 16-row tiles), per channel N = 32, K = 64
//   4 waves/block: wave -> (c = w>>1, ntile = w&1)
// =============================================================
__global__ __launch_bounds__(128) void chan_gemm(
    const float* __restrict__ all_emb, const float* __restrict__ weights /*[2][64][32]*/,
    const float* __restrict__ biases /*[2][32]*/, float* __restrict__ z /*[2][NN][32]*/)
{
    __shared__ float Xs[16 * 68];                  // 16 rows x 64, +4 pad
    const int row0 = blockIdx.x * 16;
    const int tid = threadIdx.x;

    for (int i = tid; i < 16 * 16; i += 128) {
        const int r = i >> 4, q = i & 15;
        float4 v = ((const float4*)(all_emb + (size_t)(row0 + r) * 64))[q];
        ((float4*)(Xs + r * 68))[q] = v;
    }
    __syncthreads();

    const int wave  = tid >> 5;
    const int c     = wave >> 1;
    const int ntile = wave & 1;
    const int lane  = tid & 31;
    const int mrow  = lane & 15;
    const int khalf = (lane >> 4) * 2;
    const int col   = ntile * 16 + (lane & 15);
    const float* Wc = weights + c * 64 * 32;

    v8f acc = {0.f, 0.f, 0.f, 0.f, 0.f, 0.f, 0.f, 0.f};
    #pragma unroll 4
    for (int k0 = 0; k0 < 64; k0 += 4) {
        v2f a, bb;
        a[0]  = Xs[mrow * 68 + k0 + khalf + 0];
        a[1]  = Xs[mrow * 68 + k0 + khalf + 1];
        bb[0] = Wc[(k0 + khalf + 0) * 32 + col];
        bb[1] = Wc[(k0 + khalf + 1) * 32 + col];
        acc = __builtin_amdgcn_wmma_f32_16x16x4_f32(false, a, false, bb,
                                                    (short)0, acc, false, false);
    }

    const float bias = biases[c * 32 + col];
    const int mbase = row0 + (lane >> 4) * 8;
    #pragma unroll
    for (int j = 0; j < 8; ++j)
        z[((size_t)c * NN + (mbase + j)) * 32 + col] = acc[j] + bias;
}

// =============================================================
// L2 normalize 32-float rows; one wave32 per row, lane = dim.
// out may alias in (row-independent).
// =============================================================
__global__ __launch_bounds__(256) void l2norm_rows(
    const float* __restrict__ in, float* __restrict__ out, int nrows)
{
    const int wid  = blockIdx.x * 8 + (threadIdx.x >> 5);
    const int lane = threadIdx.x & 31;
    if (wid >= nrows) return;
    float x = in[(size_t)wid * 32 + lane];
    float s = x * x;
    #pragma unroll
    for (int off = 16; off >= 1; off >>= 1) s += __shfl_xor(s, off, 32);
    const float r = rsqrtf(fmaxf(s, 1e-12f));
    out[(size_t)wid * 32 + lane] = x * r;
}

// =============================================================
// Zero fill
// =============================================================
__global__ __launch_bounds__(256) void zero_f32(float* __restrict__ p, int n)
{
    const int t = blockIdx.x * 256 + threadIdx.x;
    if (t < n) p[t] = 0.0f;
}

// =============================================================
// Per-edge: scores (2 channels), softmax over channels,
// att stored, atomic segment-sum of att into denom[c][row].
// z rows are L2-resident (76.8 MB < 192 MB L2).
// =============================================================
__global__ __launch_bounds__(256) void edge_scores(
    const float* __restrict__ z, const int* __restrict__ row_idx,
    const int* __restrict__ col_idx, float* __restrict__ att,
    float* __restrict__ denom)
{
    const int e = blockIdx.x * 256 + threadIdx.x;
    if (e >= NE) return;
    const int r = row_idx[e], cn = col_idx[e];
    float s[2];
    #pragma unroll
    for (int c = 0; c < 2; ++c) {
        const float4* a = (const float4*)(z + ((size_t)c * NN + r)  * 32);
        const float4* b = (const float4*)(z + ((size_t)c * NN + cn) * 32);
        float acc = 0.f;
        #pragma unroll
        for (int q = 0; q < 8; ++q) {
            const float4 av = a[q], bv = b[q];
            acc += av.x * bv.x + av.y * bv.y + av.z * bv.z + av.w * bv.w;
        }
        s[c] = acc;
    }
    const float m  = fmaxf(s[0], s[1]);
    const float e0 = __expf(s[0] - m);
    const float e1 = __expf(s[1] - m);
    const float inv = 1.0f / (e0 + e1);
    const float a0 = e0 * inv, a1 = e1 * inv;
    att[e]      = a0;
    att[NE + e] = a1;
    atomicAdd(&denom[r],      a0);
    atomicAdd(&denom[NN + r], a1);
}

// =============================================================
// Per (edge,dim): out[c][row][k] += (att/denom) * z[c][col][k]
// Wave layout: lanes = 32 dims of one edge -> coalesced L2
// loads of dst row and coalesced atomics into out row.
// =============================================================
__global__ __launch_bounds__(256) void edge_scatter(
    const float* __restrict__ z, const int* __restrict__ row_idx,
    const int* __restrict__ col_idx, const float* __restrict__ att,
    const float* __restrict__ denom, float* __restrict__ outz)
{
    const int e = blockIdx.x * 8 + (threadIdx.x >> 5);
    const int k = threadIdx.x & 31;
    if (e >= NE) return;
    const int r = row_idx[e], cn = col_idx[e];
    #pragma unroll
    for (int c = 0; c < 2; ++c) {
        const float na = att[(size_t)c * NE + e] /
                         fmaxf(denom[c * NN + r], 1e-12f);
        const float d = z[((size_t)c * NN + cn) * 32 + k];
        atomicAdd(&outz[((size_t)c * NN + r) * 32 + k], na * d);
    }
}

// =============================================================
// Final: out[n, c*32+k] = z[c][n][k]
// =============================================================
__global__ __launch_bounds__(256) void write_out(
    const float* __restrict__ z, float* __restrict__ outp)
{
    const int t = blockIdx.x * 256 + threadIdx.x;
    if (t >= NN * 64) return;
    const int n = t >> 6;
    const int d = t & 63;
    const int c = d >> 5;
    const int k = d & 31;
    outp[t] = z[((size_t)c * NN + n) * 32 + k];
}

// =============================================================
// Host launcher
// =============================================================
extern "C" void kernel_launch(void* const* d_in, const int* in_sizes, int n_in,
                              void* d_out, int out_size, void* d_ws, size_t ws_size,
                              hipStream_t stream)
{
    const float* user_emb  = (const float*)d_in[0];
    const float* item_emb  = (const float*)d_in[1];
    const float* user_pref = (const float*)d_in[2];
    const float* W_merge   = (const float*)d_in[3];
    const float* b_merge   = (const float*)d_in[4];
    const float* weights   = (const float*)d_in[5];
    const float* biases    = (const float*)d_in[6];
    const int*   row_idx   = (const int*)d_in[7];
    const int*   col_idx   = (const int*)d_in[8];
    float* out = (float*)d_out;

    // workspace layout (floats): allEmb | zA | zB | att | denom  (~249 MB)
    float* ws     = (float*)d_ws;
    float* allEmb = ws;                                   // NN*64
    float* zA     = allEmb + (size_t)NN * 64;             // 2*NN*32
    float* zB     = zA     + (size_t)2 * NN * 32;         // 2*NN*32
    float* att    = zB     + (size_t)2 * NN * 32;         // 2*NE
    float* denom  = att    + (size_t)2 * NE;              // 2*NN

    // merge GEMM (WMMA f32) + item copy -> all_emb
    merge_gemm<<<NU / 16, 128, 0, stream>>>(user_pref, user_emb, W_merge, b_merge, allEmb);
    copy_items<<<(NI * 16 + 255) / 256, 256, 0, stream>>>(item_emb, allEmb + (size_t)NU * 64);

    // channel projection GEMM (WMMA f32) + l2norm -> zA
    chan_gemm<<<NN / 16, 128, 0, stream>>>(allEmb, weights, biases, zA);
    l2norm_rows<<<(2 * NN) / 8, 256, 0, stream>>>(zA, zA, 2 * NN);

    for (int it = 0; it < ITERS; ++it) {
        zero_f32<<<(2 * NN + 255) / 256, 256, 0, stream>>>(denom, 2 * NN);
        zero_f32<<<(2 * NN * 32 + 255) / 256, 256, 0, stream>>>(zB, 2 * NN * 32);
        edge_scores<<<(NE + 255) / 256, 256, 0, stream>>>(zA, row_idx, col_idx, att, denom);
        edge_scatter<<<NE / 8, 256, 0, stream>>>(zA, row_idx, col_idx, att, denom, zB);
        l2norm_rows<<<(2 * NN) / 8, 256, 0, stream>>>(zB, zA, 2 * NN);
    }

    write_out<<<(NN * 64 + 255) / 256, 256, 0, stream>>>(zA, out);
}